// TripletGNN_31628139167794
// MI455X (gfx1250) — compile-verified
//
#include <hip/hip_runtime.h>

typedef __attribute__((ext_vector_type(2))) float v2f;
typedef __attribute__((ext_vector_type(8))) float v8f;

#define NN 50000
#define NE 800000

// ---------------- small elementwise kernels ----------------

__global__ void k_fill(float* p, float v, int n) {
  int i = blockIdx.x * blockDim.x + threadIdx.x;
  if (i < n) p[i] = v;
}

__global__ void k_accum_deg(const int* __restrict__ col, const float* __restrict__ w,
                            float* deg, int e) {
  int i = blockIdx.x * blockDim.x + threadIdx.x;
  if (i < e) atomicAdd(&deg[col[i]], w[i]);
}

__global__ void k_dinv(float* deg, int n) {
  int i = blockIdx.x * blockDim.x + threadIdx.x;
  if (i < n) { float d = deg[i]; deg[i] = (d > 0.f) ? rsqrtf(d) : 0.f; }
}

__global__ void k_norm(const int* __restrict__ row, const int* __restrict__ col,
                       const float* __restrict__ w, const float* __restrict__ dinv,
                       float* norm, int e) {
  int i = blockIdx.x * blockDim.x + threadIdx.x;
  if (i < e) norm[i] = dinv[row[i]] * w[i] * dinv[col[i]];
}

// Zero-pad weight matrix: W [Ks x Ns] row-major -> Wp [Kp x Np]
__global__ void k_pad(const float* __restrict__ W, float* __restrict__ Wp,
                      int Ks, int Ns, int Np, int total) {
  int i = blockIdx.x * blockDim.x + threadIdx.x;
  if (i >= total) return;
  int k = i / Np, n = i - k * Np;
  Wp[i] = (k < Ks && n < Ns) ? W[(size_t)k * Ns + n] : 0.f;
}

// A[node][j] = bias[j] + dinv[node]^2 * H[node][j]  (j < D), 0 in padding cols.
// Folds the self-loop edge (norm = dinv^2, row == col) into the init.
// One thread per float4; lda % 4 == 0, rows 16B-aligned by construction.
__global__ void k_init_agg4(const float* __restrict__ H, const float* __restrict__ dinv,
                            const float* __restrict__ bias, float* __restrict__ A,
                            int n, int D, int ldh, int lda) {
  int i = blockIdx.x * blockDim.x + threadIdx.x;
  int q = lda >> 2;
  int node = i / q, jc = i - node * q;
  if (node >= n) return;
  int j0 = jc * 4;
  float di = dinv[node];
  float d2 = di * di;
  float4 h = *(const float4*)(H + (size_t)node * ldh + j0);
  float4 o;
  o.x = (j0 + 0 < D) ? bias[j0 + 0] + d2 * h.x : 0.f;
  o.y = (j0 + 1 < D) ? bias[j0 + 1] + d2 * h.y : 0.f;
  o.z = (j0 + 2 < D) ? bias[j0 + 2] + d2 * h.z : 0.f;
  o.w = (j0 + 3 < D) ? bias[j0 + 3] + d2 * h.w : 0.f;
  *(float4*)(A + (size_t)node * lda + j0) = o;
}

// Edge scatter: A[col[e]][j] += norm[e] * H[row[e]][j]
// One thread per (edge, 4-col chunk); float4 gather (global_load_b128),
// scalar f32 atomics into L2-resident accumulator.
__global__ void k_scatter(const int* __restrict__ row, const int* __restrict__ col,
                          const float* __restrict__ norm, const float* __restrict__ H,
                          float* __restrict__ A, int e, int D, int ldh, int lda, int nch) {
  int i = blockIdx.x * blockDim.x + threadIdx.x;
  int ed = i / nch, ch = i - ed * nch;
  if (ed >= e) return;
  int r = row[ed], c = col[ed];
  float s = norm[ed];
  const float* hr = H + (size_t)r * ldh;
  float* ac = A + (size_t)c * lda;
  int j0 = ch * 4;
  if (j0 + 4 <= D) {
    float4 h = *(const float4*)(hr + j0);
    atomicAdd(&ac[j0 + 0], s * h.x);
    atomicAdd(&ac[j0 + 1], s * h.y);
    atomicAdd(&ac[j0 + 2], s * h.z);
    atomicAdd(&ac[j0 + 3], s * h.w);
  } else {
    for (int j = j0; j < D; ++j) atomicAdd(&ac[j], s * hr[j]);
  }
}

// ---------------- WMMA f32 GEMM, NT-way N-blocked ----------------
// One wave computes a 16 x (NT*16) strip of C; each K-step loads one A
// fragment and NT B fragments, issuing NT v_wmma_f32_16x16x4_f32.
// A (32-bit, 16x4): lane 0-15 m=lane; VGPR v holds K = v + 2*(lane>=16)
// B (32-bit, 4x16): n = lane&15;     VGPR v holds K = v + 2*(lane>=16)
// C/D:              VGPR v -> row = v + 8*(lane>=16), col = lane&15
template <int NT>
__global__ void k_gemm_wmma(const float* __restrict__ A, const float* __restrict__ B,
                            float* __restrict__ C, int M, int Nstrips, int K,
                            int lda, int ldb, int ldc, int relu_in) {
  int gtid = blockIdx.x * blockDim.x + threadIdx.x;
  int wave = gtid >> 5;
  int lane = threadIdx.x & 31;
  int mtiles = M >> 4;
  if (wave >= mtiles * Nstrips) return;
  int tm = wave / Nstrips;
  int ts = wave - tm * Nstrips;

  int half = lane >> 4;  // 0: lanes 0-15, 1: lanes 16-31
  int l16  = lane & 15;
  int m  = tm * 16 + l16;
  int n0 = ts * (NT * 16) + l16;

  const float* arow  = A + (size_t)m * lda;
  const float* bbase = B + n0;

  v8f zero = {};
  v8f c[NT];
#pragma unroll
  for (int t = 0; t < NT; ++t) c[t] = zero;

  for (int k0 = 0; k0 < K; k0 += 4) {
    int ka = k0 + 2 * half;
    v2f a;
    a.x = arow[ka];
    a.y = arow[ka + 1];
    if (relu_in) { a.x = fmaxf(a.x, 0.f); a.y = fmaxf(a.y, 0.f); }
    const float* brow0 = bbase + (size_t)ka * ldb;
    const float* brow1 = bbase + (size_t)(ka + 1) * ldb;
#pragma unroll
    for (int t = 0; t < NT; ++t) {
      v2f b;
      b.x = brow0[t * 16];
      b.y = brow1[t * 16];
      // (neg_a, A, neg_b, B, c_mod, C, reuse_a, reuse_b)
      c[t] = __builtin_amdgcn_wmma_f32_16x16x4_f32(false, a, false, b, (short)0,
                                                   c[t], false, false);
    }
  }

#pragma unroll
  for (int t = 0; t < NT; ++t) {
#pragma unroll
    for (int v = 0; v < 8; ++v) {
      int mr = tm * 16 + v + 8 * half;
      C[(size_t)mr * ldc + n0 + t * 16] = c[t][v];
    }
  }
}

// ---------------- host launch ----------------

static inline int cdiv(long long a, int b) { return (int)((a + b - 1) / b); }

extern "C" void kernel_launch(void* const* d_in, const int* in_sizes, int n_in,
                              void* d_out, int out_size, void* d_ws, size_t ws_size,
                              hipStream_t stream) {
  const float* x  = (const float*)d_in[0];  // [N,128]
  const int*   ei = (const int*)d_in[1];    // [2,E]
  const float* w  = (const float*)d_in[2];  // [E]
  const float* W1 = (const float*)d_in[3];  // [128,150]
  const float* b1 = (const float*)d_in[4];  // [150]
  const float* W2 = (const float*)d_in[5];  // [150,300]
  const float* b2 = (const float*)d_in[6];  // [300]
  float* out = (float*)d_out;               // [N,300]

  const int N = NN, E = NE;
  const int D0 = 128;            // in features
  const int D1 = 150, D1P = 160; // hidden (pad to 10 tiles)
  const int D2 = 300, D2P = 320; // out (pad to 20 tiles, so strips of 5 divide)
  const int* row = ei;
  const int* col = ei + E;

  // workspace carve-up (floats; every block a multiple of 16B)
  float* deg  = (float*)d_ws;             // 50048
  float* norm = deg  + 50048;             // 800000
  float* W1p  = norm + 800000;            // 128*160 = 20480
  float* W2p  = W1p  + 128 * 160;         // 160*320 = 51200
  float* H1   = W2p  + 160 * 320;         // N*160
  float* A1   = H1   + (size_t)N * D1P;   // N*160
  float* H2   = A1   + (size_t)N * D1P;   // N*320
  // total ~132 MB

  const int T = 256;

  // ---- gcn_norm ----
  k_fill<<<cdiv(N, T), T, 0, stream>>>(deg, 1.0f, N);  // self-loop weight
  k_accum_deg<<<cdiv(E, T), T, 0, stream>>>(col, w, deg, E);
  k_dinv<<<cdiv(N, T), T, 0, stream>>>(deg, N);        // deg -> dinv in place
  k_norm<<<cdiv(E, T), T, 0, stream>>>(row, col, w, deg, norm, E);

  // ---- pad weights ----
  k_pad<<<cdiv(128 * 160, T), T, 0, stream>>>(W1, W1p, D0, D1, D1P, 128 * 160);
  k_pad<<<cdiv(160 * 320, T), T, 0, stream>>>(W2, W2p, D1, D2, D2P, 160 * 320);

  // ---- layer 1: H1 = X @ W1p  (strips of 5 N-tiles -> 2 strips) ----
  {
    int strips = (D1P / 16) / 5;  // 2
    long long waves = (long long)(N / 16) * strips;
    k_gemm_wmma<5><<<cdiv(waves * 32, T), T, 0, stream>>>(
        x, W1p, H1, N, strips, D0, D0, D1P, D1P, 0);
  }
  // A1 = b1 + dinv^2 * H1 (self-loop term), zero padding cols
  k_init_agg4<<<cdiv((long long)N * D1P / 4, T), T, 0, stream>>>(
      H1, deg, b1, A1, N, D1, D1P, D1P);
  {
    int nch = (D1 + 3) / 4;  // 38
    k_scatter<<<cdiv((long long)E * nch, T), T, 0, stream>>>(
        row, col, norm, H1, A1, E, D1, D1P, D1P, nch);
  }

  // ---- layer 2: H2 = relu(A1) @ W2p  (ReLU fused into A-load; 4 strips) ----
  {
    int strips = (D2P / 16) / 5;  // 4
    long long waves = (long long)(N / 16) * strips;
    k_gemm_wmma<5><<<cdiv(waves * 32, T), T, 0, stream>>>(
        A1, W2p, H2, N, strips, D1P, D1P, D2P, D2P, 1);
  }
  // out = b2 + dinv^2 * H2 (self-loop term); full overwrite of d_out
  k_init_agg4<<<cdiv((long long)N * D2 / 4, T), T, 0, stream>>>(
      H2, deg, b2, out, N, D2, D2P, D2);
  {
    int nch = D2 / 4;  // 75
    k_scatter<<<cdiv((long long)E * nch, T), T, 0, stream>>>(
        row, col, norm, H2, out, E, D2, D2P, D2, nch);
  }
}